// MarginDevianceLoss_49503793054565
// MI455X (gfx1250) — compile-verified
//
#include <hip/hip_runtime.h>
#include <hip/hip_bf16.h>

#define NROW 4096
#define DDIM 128
#define KPOS 7
#define NNEG 4088.0f

typedef float v2f __attribute__((ext_vector_type(2)));
typedef float v8f __attribute__((ext_vector_type(8)));

__device__ __forceinline__ float softplus_fast(float x) {
  // log1p(exp(x)), stable; e in (0,1] so __logf(1+e) is accurate enough here
  float e = __expf(-fabsf(x));
  return fmaxf(x, 0.0f) + __logf(1.0f + e);
}

__device__ __forceinline__ v8f wmma_f32(v2f a, v2f b, v8f c) {
  return __builtin_amdgcn_wmma_f32_16x16x4_f32(false, a, false, b,
                                               (short)0, c, false, false);
}

// ---------------------------------------------------------------------------
// K1: sim = X X^T via V_WMMA_F32_16X16X4_F32. Each wave holds the 16x128 A
// fragment in registers and processes TWO 16x16 n-tiles concurrently, each
// with even/odd K accumulation chains (4 independent WMMA chains for ILP).
// Streams per-row sum / sumsq; stashes the 16x16 diagonal tile to workspace.
// grid = 256 blocks (one 16-row strip each), 256 threads = 8 waves.
// ---------------------------------------------------------------------------
__global__ void __launch_bounds__(256) k_gemm_stats(const float* __restrict__ X,
                                                    float* __restrict__ row_sum,
                                                    float* __restrict__ row_sq,
                                                    float* __restrict__ diag) {
  const int lane = threadIdx.x & 31;
  const int wave = threadIdx.x >> 5;
  const int l16  = lane & 15;
  const int half = lane >> 4;
  const int m0   = blockIdx.x << 4;

  const float* Ap = X + (size_t)(m0 + l16) * DDIM + half * 2;
  v2f a[32];
#pragma unroll
  for (int ks = 0; ks < 32; ++ks) a[ks] = *(const v2f*)(Ap + ks * 4);

  float sacc[8], qacc[8];
#pragma unroll
  for (int v = 0; v < 8; ++v) { sacc[v] = 0.0f; qacc[v] = 0.0f; }

  const v8f vzero = {0.f, 0.f, 0.f, 0.f, 0.f, 0.f, 0.f, 0.f};

  for (int j = 0; j < 16; ++j) {           // 32 tiles per wave, 2 at a time
    const int ta  = wave + j * 16;         // tile indices ta, ta+8
    const int n0a = ta << 4;
    const int n0b = n0a + 128;
    const float* Bpa = X + (size_t)(n0a + l16) * DDIM + half * 2;
    const float* Bpb = X + (size_t)(n0b + l16) * DDIM + half * 2;

    v8f c0 = vzero, c1 = vzero, d0 = vzero, d1 = vzero;
#pragma unroll
    for (int ks = 0; ks < 32; ks += 2) {   // 4 independent WMMA chains
      v2f ba0 = *(const v2f*)(Bpa + ks * 4);
      v2f bb0 = *(const v2f*)(Bpb + ks * 4);
      v2f ba1 = *(const v2f*)(Bpa + ks * 4 + 4);
      v2f bb1 = *(const v2f*)(Bpb + ks * 4 + 4);
      c0 = wmma_f32(a[ks],     ba0, c0);
      d0 = wmma_f32(a[ks],     bb0, d0);
      c1 = wmma_f32(a[ks + 1], ba1, c1);
      d1 = wmma_f32(a[ks + 1], bb1, d1);
    }
    v8f ca = c0 + c1;
    v8f cb = d0 + d1;

#pragma unroll
    for (int v = 0; v < 8; ++v) {
      float x = ca[v], y = cb[v];
      sacc[v] += x + y;
      qacc[v] += x * x + y * y;
    }
    if (n0a == m0) {
#pragma unroll
      for (int v = 0; v < 8; ++v)
        diag[(blockIdx.x << 8) + ((v + (half << 3)) << 4) + l16] = ca[v];
    }
    if (n0b == m0) {
#pragma unroll
      for (int v = 0; v < 8; ++v)
        diag[(blockIdx.x << 8) + ((v + (half << 3)) << 4) + l16] = cb[v];
    }
  }

  __shared__ float ls[8][16];
  __shared__ float lq[8][16];
#pragma unroll
  for (int v = 0; v < 8; ++v) {
    float s = sacc[v], q = qacc[v];
#pragma unroll
    for (int m = 1; m < 16; m <<= 1) {
      s += __shfl_xor(s, m, 32);
      q += __shfl_xor(q, m, 32);
    }
    if (l16 == 0) { ls[wave][v + (half << 3)] = s; lq[wave][v + (half << 3)] = q; }
  }
  __syncthreads();
  if (threadIdx.x < 16) {
    float s = 0.0f, q = 0.0f;
#pragma unroll
    for (int w = 0; w < 8; ++w) { s += ls[w][threadIdx.x]; q += lq[w][threadIdx.x]; }
    row_sum[m0 + threadIdx.x] = s;
    row_sq[m0 + threadIdx.x]  = q;
  }
}

// ---------------------------------------------------------------------------
// K2: per-row finalize pos/neg stats -> inter, thr = min(pos)-0.05, pos_loss.
// ---------------------------------------------------------------------------
__global__ void __launch_bounds__(256) k_rowstats(const float* __restrict__ diag,
                                                  const float* __restrict__ row_sum,
                                                  const float* __restrict__ row_sq,
                                                  float* __restrict__ inter_o,
                                                  float* __restrict__ thr_o,
                                                  float* __restrict__ ploss_o) {
  int i = blockIdx.x * 256 + threadIdx.x;
  if (i >= NROW) return;
  int s = i >> 4, r = i & 15, h = r >> 3, self = r & 7;
  const float* dp = diag + (s << 8) + (r << 4) + (h << 3);  // this row's 8 group cols
  float pv[8], sum8 = 0.0f, sq8 = 0.0f;
#pragma unroll
  for (int c = 0; c < 8; ++c) { float x = dp[c]; pv[c] = x; sum8 += x; sq8 += x * x; }
  float ds   = pv[self];
  float psum = sum8 - ds, psq = sq8 - ds * ds;
  float pmean = psum * (1.0f / KPOS);
  float pstd  = sqrtf(fmaxf(psq * (1.0f / KPOS) - pmean * pmean, 0.0f));
  float pmin = 1e30f;
#pragma unroll
  for (int c = 0; c < 8; ++c) if (c != self) pmin = fminf(pmin, pv[c]);
  float nsum = row_sum[i] - sum8, nsq = row_sq[i] - sq8;
  float nmean = nsum * (1.0f / NNEG);
  float nstd  = sqrtf(fmaxf(nsq * (1.0f / NNEG) - nmean * nmean, 0.0f));
  float it = (nstd * pmean + pstd * nmean) / (pstd + nstd);
  it = 0.8f * it + 0.1f;
  float pl = 0.0f;
#pragma unroll
  for (int c = 0; c < 8; ++c)
    if (c != self) pl += softplus_fast(-10.0f * (pv[c] - it));
  ploss_o[i] = pl * (0.2f / KPOS);
  inter_o[i] = it;
  thr_o[i]   = pmin - 0.05f;
}

// ---------------------------------------------------------------------------
// K3: recompute sim tiles (same 2-tile / 4-chain WMMA skeleton), accumulate
// per-row kept-negative count and sum of softplus(40*(x - inter)), masking
// the 8 group columns (they only occur in the diagonal tile).
// ---------------------------------------------------------------------------
__global__ void __launch_bounds__(256) k_gemm_negloss(const float* __restrict__ X,
                                                      const float* __restrict__ inter_i,
                                                      const float* __restrict__ thr_i,
                                                      float* __restrict__ ncnt,
                                                      float* __restrict__ nval) {
  const int lane = threadIdx.x & 31;
  const int wave = threadIdx.x >> 5;
  const int l16  = lane & 15;
  const int half = lane >> 4;
  const int m0   = blockIdx.x << 4;

  const float* Ap = X + (size_t)(m0 + l16) * DDIM + half * 2;
  v2f a[32];
#pragma unroll
  for (int ks = 0; ks < 32; ++ks) a[ks] = *(const v2f*)(Ap + ks * 4);

  float itv[8], thv[8];
#pragma unroll
  for (int v = 0; v < 8; ++v) {
    int row = m0 + v + (half << 3);
    itv[v] = inter_i[row];
    thv[v] = thr_i[row];
  }

  float cacc[8], vacc[8];
#pragma unroll
  for (int v = 0; v < 8; ++v) { cacc[v] = 0.0f; vacc[v] = 0.0f; }

  const bool ingrp = ((l16 >> 3) == half);  // column in this row's group (diag tile only)
  const v8f vzero = {0.f, 0.f, 0.f, 0.f, 0.f, 0.f, 0.f, 0.f};

  for (int j = 0; j < 16; ++j) {
    const int ta  = wave + j * 16;
    const int n0a = ta << 4;
    const int n0b = n0a + 128;
    const float* Bpa = X + (size_t)(n0a + l16) * DDIM + half * 2;
    const float* Bpb = X + (size_t)(n0b + l16) * DDIM + half * 2;

    v8f c0 = vzero, c1 = vzero, d0 = vzero, d1 = vzero;
#pragma unroll
    for (int ks = 0; ks < 32; ks += 2) {
      v2f ba0 = *(const v2f*)(Bpa + ks * 4);
      v2f bb0 = *(const v2f*)(Bpb + ks * 4);
      v2f ba1 = *(const v2f*)(Bpa + ks * 4 + 4);
      v2f bb1 = *(const v2f*)(Bpb + ks * 4 + 4);
      c0 = wmma_f32(a[ks],     ba0, c0);
      d0 = wmma_f32(a[ks],     bb0, d0);
      c1 = wmma_f32(a[ks + 1], ba1, c1);
      d1 = wmma_f32(a[ks + 1], bb1, d1);
    }
    v8f ca = c0 + c1;
    v8f cb = d0 + d1;

    const float excla = ((n0a == m0) && ingrp) ? 0.0f : 1.0f;
    const float exclb = ((n0b == m0) && ingrp) ? 0.0f : 1.0f;
#pragma unroll
    for (int v = 0; v < 8; ++v) {
      float x = ca[v];
      float keep = (x > thv[v]) ? excla : 0.0f;
      cacc[v] += keep;
      vacc[v] += keep * softplus_fast(40.0f * (x - itv[v]));
      float y = cb[v];
      float keepb = (y > thv[v]) ? exclb : 0.0f;
      cacc[v] += keepb;
      vacc[v] += keepb * softplus_fast(40.0f * (y - itv[v]));
    }
  }

  __shared__ float lc[8][16];
  __shared__ float lv[8][16];
#pragma unroll
  for (int v = 0; v < 8; ++v) {
    float cC = cacc[v], vV = vacc[v];
#pragma unroll
    for (int m = 1; m < 16; m <<= 1) {
      cC += __shfl_xor(cC, m, 32);
      vV += __shfl_xor(vV, m, 32);
    }
    if (l16 == 0) { lc[wave][v + (half << 3)] = cC; lv[wave][v + (half << 3)] = vV; }
  }
  __syncthreads();
  if (threadIdx.x < 16) {
    float cC = 0.0f, vV = 0.0f;
#pragma unroll
    for (int w = 0; w < 8; ++w) { cC += lc[w][threadIdx.x]; vV += lv[w][threadIdx.x]; }
    ncnt[m0 + threadIdx.x] = cC;
    nval[m0 + threadIdx.x] = vV;
  }
}

// ---------------------------------------------------------------------------
// K4: final scalar reduction.
// ---------------------------------------------------------------------------
__global__ void __launch_bounds__(256) k_final(const float* __restrict__ ploss,
                                               const float* __restrict__ ncnt,
                                               const float* __restrict__ nval,
                                               float* __restrict__ out) {
  __shared__ float red[256];
  float acc = 0.0f;
  for (int i = threadIdx.x; i < NROW; i += 256)
    acc += ploss[i] + 0.05f * nval[i] / fmaxf(ncnt[i], 1.0f);
  red[threadIdx.x] = acc;
  __syncthreads();
  for (int s = 128; s > 0; s >>= 1) {
    if (threadIdx.x < s) red[threadIdx.x] += red[threadIdx.x + s];
    __syncthreads();
  }
  if (threadIdx.x == 0) out[0] = red[0] * (1.0f / NROW);
}

extern "C" void kernel_launch(void* const* d_in, const int* in_sizes, int n_in,
                              void* d_out, int out_size, void* d_ws, size_t ws_size,
                              hipStream_t stream) {
  (void)in_sizes; (void)n_in; (void)out_size; (void)ws_size;
  const float* X = (const float*)d_in[0];   // [4096,128] L2-normalized fp32
  float* out = (float*)d_out;               // scalar fp32

  float* ws       = (float*)d_ws;
  float* row_sum  = ws;                     // 4096
  float* row_sq   = ws + 4096;              // 4096
  float* diag     = ws + 8192;              // 256 * 256 = 65536
  float* inter    = diag + 65536;           // 4096
  float* thr      = inter + 4096;           // 4096
  float* ploss    = thr + 4096;             // 4096
  float* ncnt     = ploss + 4096;           // 4096
  float* nval     = ncnt + 4096;            // 4096  (total ~368 KB)

  k_gemm_stats<<<256, 256, 0, stream>>>(X, row_sum, row_sq, diag);
  k_rowstats<<<16, 256, 0, stream>>>(diag, row_sum, row_sq, inter, thr, ploss);
  k_gemm_negloss<<<256, 256, 0, stream>>>(X, inter, thr, ncnt, nval);
  k_final<<<1, 256, 0, stream>>>(ploss, ncnt, nval, out);
}